// GNNRNNv2_27307402068444
// MI455X (gfx1250) — compile-verified
//
#include <hip/hip_runtime.h>
#include <hip/hip_bf16.h>

// ---------------------------------------------------------------------------
// GNN-RNN (encoder -> GAT -> per-feature GRU -> decoder) for gfx1250.
// Matmuls use V_WMMA_F32_16X16X4_F32 (full fp32, 16x16 C/D tiles).
// GRU input tiles are double-buffered into LDS with
// GLOBAL_LOAD_ASYNC_TO_LDS_B128 (ASYNCcnt) to hide latency behind WMMAs.
// B=16, T=64, F=N=64, H=64, HEADS=4, D=16, 3H=192.
// ---------------------------------------------------------------------------

typedef float v2f __attribute__((ext_vector_type(2)));
typedef float v8f __attribute__((ext_vector_type(8)));

__device__ __forceinline__ v8f wmma4(v2f a, v2f b, v8f c) {
    // D = A(16x4, f32) * B(4x16, f32) + C(16x16, f32)
    return __builtin_amdgcn_wmma_f32_16x16x4_f32(
        false, a, false, b, (short)0, c, false, false);
}

__device__ __forceinline__ void async_ld_b128(unsigned lds_addr, const void* gaddr) {
    // GV mode: per-lane 64-bit global address, per-lane LDS byte address.
    asm volatile("global_load_async_to_lds_b128 %0, %1, off"
                 :: "v"(lds_addr), "v"(gaddr) : "memory");
}

__device__ __forceinline__ void wait_async0() {
    asm volatile("s_wait_asynccnt 0x0" ::: "memory");
}

// ---------------------------------------------------------------------------
// Kernel 1: deterministic CSR build (edges sorted by dst, stable in edge order)
// ---------------------------------------------------------------------------
__global__ void csr_kernel(const int* __restrict__ src, const int* __restrict__ dst,
                           int E, int* __restrict__ offs, int* __restrict__ srcs)
{
    __shared__ int cnt[64];
    const int tid = threadIdx.x;   // 64 threads, one per dst node
    if (tid < 64) {
        int c = 0;
        for (int e = 0; e < E; ++e) c += (dst[e] == tid);
        cnt[tid] = c;
    }
    __syncthreads();
    if (tid == 0) {
        int acc = 0;
        for (int n = 0; n < 64; ++n) { offs[n] = acc; acc += cnt[n]; }
        offs[64] = acc;
    }
    __syncthreads();
    if (tid < 64) {
        int base = 0;
        for (int n = 0; n < tid; ++n) base += cnt[n];
        int p = base;
        for (int e = 0; e < E; ++e)
            if (dst[e] == tid) srcs[p++] = src[e];
    }
}

// ---------------------------------------------------------------------------
// Kernel 2: encoder + GAT. One block per (b,t) graph. 256 threads = 8 waves.
// ---------------------------------------------------------------------------
__global__ __launch_bounds__(256) void gat_kernel(
    const float* __restrict__ hist, const float* __restrict__ W_enc,
    const float* __restrict__ b_enc, const float* __restrict__ W_gat,
    const float* __restrict__ attn_l, const float* __restrict__ attn_r,
    const float* __restrict__ b_gat, const int* __restrict__ offs,
    const int* __restrict__ srcs, float* __restrict__ g_out)
{
    __shared__ float xs[64 * 64];     // encoded node features
    __shared__ float ft[64 * 64];     // GAT fc output (feat)
    __shared__ float elL[64 * 4];
    __shared__ float erL[64 * 4];
    __shared__ float histL[64];

    const int tid  = threadIdx.x;
    const int lane = tid & 31;
    const int w    = tid >> 5;
    const int bt   = blockIdx.x;              // b*T + t
    const int m16  = lane & 15;
    const int koff = (lane >> 4) << 1;        // 0 or 2
    const int cr   = (lane >> 4) << 3;        // 0 or 8

    if (tid < 64) histL[tid] = hist[bt * 64 + tid];
    __syncthreads();

    // encoder: Linear(1,H) per node
    for (int i = tid; i < 64 * 64; i += 256) {
        const int n = i >> 6;
        xs[i] = histL[n] * W_enc[i] + b_enc[i];
    }
    __syncthreads();

    // feat = xs @ W_gat : 4x4 tiles of 16x16, K=64 -> 16 wmma steps per tile
    for (int rep = 0; rep < 2; ++rep) {
        const int tile = w + (rep << 3);      // 0..15
        const int mt = tile >> 2, nt = tile & 3;
        const float* arow = &xs[(mt * 16 + m16) * 64];
        const int col = nt * 16 + m16;
        v8f acc = {};
        #pragma unroll
        for (int kt = 0; kt < 16; ++kt) {
            const int k0 = (kt << 2) + koff;
            v2f a = { arow[k0], arow[k0 + 1] };
            v2f b = { W_gat[k0 * 64 + col], W_gat[(k0 + 1) * 64 + col] };
            acc = wmma4(a, b, acc);
        }
        #pragma unroll
        for (int r = 0; r < 8; ++r)
            ft[(mt * 16 + cr + r) * 64 + col] = acc[r];
    }
    __syncthreads();

    // el/er: one thread per (node, head)
    {
        const int n = tid >> 2, hd = tid & 3;
        float sl = 0.f, sr = 0.f;
        #pragma unroll
        for (int d = 0; d < 16; ++d) {
            const float fv = ft[n * 64 + hd * 16 + d];
            sl += fv * attn_l[hd * 16 + d];
            sr += fv * attn_r[hd * 16 + d];
        }
        elL[tid] = sl;   // tid == n*4 + hd
        erL[tid] = sr;
    }
    __syncthreads();

    // softmax + aggregate: one thread per (dst, head)
    {
        const int dn = tid >> 2, hd = tid & 3;
        const int e0 = offs[dn], e1 = offs[dn + 1];
        const float er_d = erL[dn * 4 + hd];
        float mx = -3.4e38f;
        for (int e = e0; e < e1; ++e) {
            float v = elL[srcs[e] * 4 + hd] + er_d;
            v = (v > 0.f) ? v : 0.2f * v;          // leaky_relu
            mx = fmaxf(mx, v);
        }
        float den = 0.f;
        float msg[16];
        #pragma unroll
        for (int d = 0; d < 16; ++d) msg[d] = 0.f;
        for (int e = e0; e < e1; ++e) {
            const int s = srcs[e];
            float v = elL[s * 4 + hd] + er_d;
            v = (v > 0.f) ? v : 0.2f * v;
            const float ex = __expf(v - mx);
            den += ex;
            const float* fr = &ft[s * 64 + hd * 16];
            #pragma unroll
            for (int d = 0; d < 16; ++d) msg[d] += ex * fr[d];
        }
        const float inv = 1.f / den;
        float* go = &g_out[(bt * 64 + dn) * 64 + hd * 16];
        #pragma unroll
        for (int d = 0; d < 16; ++d)
            go[d] = msg[d] * inv + b_gat[hd * 16 + d];
    }
}

// ---------------------------------------------------------------------------
// Kernel 3: per-feature GRU. One block per feature f. 384 thr = 12 waves.
// Waves 0-3: r gate, 4-7: z gate (combined x+h projections); 8-11: n gate
// (x and h kept separate). W_ih/W_hh B-fragments stay in VGPRs for the whole
// T=64 scan; h lives in LDS. x_{t+1} is async-DMA'd into a double-buffered
// LDS tile while step t computes. hid[] is read (g) then overwritten (h).
// ---------------------------------------------------------------------------
__global__ __launch_bounds__(384) void gru_kernel(
    float* __restrict__ hid,              // [B][T][F][H] : in g, out hiddens
    const float* __restrict__ W_ih,       // [F][192][64]
    const float* __restrict__ W_hh,       // [F][192][64]
    const float* __restrict__ b_ih,       // [F][192]
    const float* __restrict__ b_hh)       // [F][192]
{
    __shared__ float hL[16 * 64];
    __shared__ float rL[16 * 64];
    __shared__ float zL[16 * 64];
    __shared__ float xL[2][16 * 64];      // double-buffered x_t tile

    const int f    = blockIdx.x;
    const int tid  = threadIdx.x;
    const int lane = tid & 31;
    const int w    = tid >> 5;            // 0..11
    const int gate = w >> 2;              // 0=r, 1=z, 2=n
    const int nt   = w & 3;
    const int m16  = lane & 15;
    const int koff = (lane >> 4) << 1;
    const int cr   = (lane >> 4) << 3;
    const int col  = gate * 64 + nt * 16 + m16;   // gate column 0..191
    const int jcol = nt * 16 + m16;               // hidden column 0..63

    // async staging assignment: threads 0..255 move one 16B chunk each
    const int cm = tid >> 4;              // batch row 0..15 (tid<256)
    const int cp = (tid & 15) << 2;       // float offset 0..60 step 4

    for (int i = tid; i < 16 * 64; i += 384) hL[i] = 0.f;

    // stage x_0 into buffer 0 (overlapped with weight preloads below)
    if (tid < 256) {
        const float* g0 = &hid[((cm * 64 + 0) * 64 + f) * 64 + cp];
        async_ld_b128((unsigned)(uintptr_t)&xL[0][cm * 64 + cp], g0);
    }

    // resident B-fragments: B[k][n] = W[f][n][k] (contiguous in k)
    v2f bi[16], bh[16];
    {
        const float* wi = &W_ih[(f * 192 + col) * 64];
        const float* wh = &W_hh[(f * 192 + col) * 64];
        #pragma unroll
        for (int kt = 0; kt < 16; ++kt) {
            const int k0 = (kt << 2) + koff;
            bi[kt] = (v2f){ wi[k0], wi[k0 + 1] };
            bh[kt] = (v2f){ wh[k0], wh[k0 + 1] };
        }
    }
    const float biasI = b_ih[f * 192 + col];
    const float biasH = b_hh[f * 192 + col];
    wait_async0();
    __syncthreads();

    for (int t = 0; t < 64; ++t) {
        const int buf = t & 1;

        // kick off DMA of x_{t+1} into the other buffer
        if (t < 63 && tid < 256) {
            const float* gn = &hid[((cm * 64 + (t + 1)) * 64 + f) * 64 + cp];
            async_ld_b128((unsigned)(uintptr_t)&xL[buf ^ 1][cm * 64 + cp], gn);
        }

        // A-fragments: x_t and h, both from LDS
        v2f ax[16], ah[16];
        #pragma unroll
        for (int kt = 0; kt < 16; ++kt) {
            const int k0 = (kt << 2) + koff;
            ax[kt] = (v2f){ xL[buf][m16 * 64 + k0], xL[buf][m16 * 64 + k0 + 1] };
            ah[kt] = (v2f){ hL[m16 * 64 + k0], hL[m16 * 64 + k0 + 1] };
        }

        v8f accX = {}, accH = {};
        if (gate < 2) {
            const float bb = biasI + biasH;
            #pragma unroll
            for (int r = 0; r < 8; ++r) accX[r] = bb;
            #pragma unroll
            for (int kt = 0; kt < 16; ++kt) {
                accX = wmma4(ax[kt], bi[kt], accX);   // x @ Wih^T
                accX = wmma4(ah[kt], bh[kt], accX);   // h @ Whh^T
            }
            float* dstL = (gate == 0) ? rL : zL;
            #pragma unroll
            for (int r = 0; r < 8; ++r) {
                const float v = 1.f / (1.f + __expf(-accX[r]));  // sigmoid
                dstL[(cr + r) * 64 + jcol] = v;
            }
        } else {
            #pragma unroll
            for (int r = 0; r < 8; ++r) { accX[r] = biasI; accH[r] = biasH; }
            #pragma unroll
            for (int kt = 0; kt < 16; ++kt) {
                accX = wmma4(ax[kt], bi[kt], accX);   // xn
                accH = wmma4(ah[kt], bh[kt], accH);   // hn
            }
        }
        __syncthreads();   // r,z visible; all xL[buf] reads complete

        if (gate == 2) {
            #pragma unroll
            for (int r = 0; r < 8; ++r) {
                const int m = cr + r;
                const float rv = rL[m * 64 + jcol];
                const float zv = zL[m * 64 + jcol];
                const float hp = hL[m * 64 + jcol];
                const float nv = tanhf(accX[r] + rv * accH[r]);
                const float hn = (1.f - zv) * nv + zv * hp;
                hL[m * 64 + jcol] = hn;
                hid[((m * 64 + t) * 64 + f) * 64 + jcol] = hn;  // hiddens out
            }
        }
        wait_async0();     // x_{t+1} landed in LDS (per-wave ASYNCcnt)
        __syncthreads();   // h_t and x_{t+1} visible to all waves
    }
}

// ---------------------------------------------------------------------------
// Kernel 4: decoder. ans[b,t,f] = hiddens[b,t,f,:] . W_dec[f,:] + b_dec[f].
// One thread per (b,t,f); fully parallel, L2-resident.
// ---------------------------------------------------------------------------
__global__ __launch_bounds__(256) void dec_kernel(
    const float* __restrict__ hid, const float* __restrict__ W_dec,
    const float* __restrict__ b_dec, float* __restrict__ ans)
{
    const int i = blockIdx.x * 256 + threadIdx.x;   // (b*T + t)*F + f
    const int f = i & 63;
    const float* h = &hid[i * 64];
    const float* wv = &W_dec[f * 64];
    float s = b_dec[f];
    #pragma unroll
    for (int c = 0; c < 64; ++c) s += h[c] * wv[c];
    ans[i] = s;
}

// ---------------------------------------------------------------------------
extern "C" void kernel_launch(void* const* d_in, const int* in_sizes, int n_in,
                              void* d_out, int out_size, void* d_ws, size_t ws_size,
                              hipStream_t stream) {
    const float* hist   = (const float*)d_in[0];
    const int*   src    = (const int*)  d_in[1];
    const int*   dst    = (const int*)  d_in[2];
    const float* W_enc  = (const float*)d_in[3];
    const float* b_enc  = (const float*)d_in[4];
    const float* W_gat  = (const float*)d_in[5];
    const float* attn_l = (const float*)d_in[6];
    const float* attn_r = (const float*)d_in[7];
    const float* b_gat  = (const float*)d_in[8];
    const float* W_ih   = (const float*)d_in[9];
    const float* W_hh   = (const float*)d_in[10];
    const float* b_ih   = (const float*)d_in[11];
    const float* b_hh   = (const float*)d_in[12];
    const float* W_dec  = (const float*)d_in[13];
    const float* b_dec  = (const float*)d_in[14];

    const int E = in_sizes[1];            // number of edges (incl. self loops)

    float* ans = (float*)d_out;           // [B][T][F]
    float* hid = ans + 16 * 64 * 64;      // [B][T][F][H]: g, then hiddens

    int* offs = (int*)d_ws;               // [65]
    int* srcs = offs + 65;                // [E]

    csr_kernel<<<1, 64, 0, stream>>>(src, dst, E, offs, srcs);
    gat_kernel<<<16 * 64, 256, 0, stream>>>(hist, W_enc, b_enc, W_gat,
                                            attn_l, attn_r, b_gat,
                                            offs, srcs, hid);
    gru_kernel<<<64, 384, 0, stream>>>(hid, W_ih, W_hh, b_ih, b_hh);
    dec_kernel<<<(16 * 64 * 64) / 256, 256, 0, stream>>>(hid, W_dec, b_dec, ans);
}